// OTPrompt_64183991271489
// MI455X (gfx1250) — compile-verified
//
#include <hip/hip_runtime.h>

typedef float v2f __attribute__((ext_vector_type(2)));
typedef float v8f __attribute__((ext_vector_type(8)));

#define N_FEATS 100
#define M_PROMPTS 20
#define DIM 128
#define NPAD 112   // 7 * 16
#define MPAD 32    // 2 * 16
#define XS 132     // x_lds row stride (dwords) -> conflict-free column reads
#define PS 132     // p_lds row stride
#define KS 33      // K_lds row stride (33 odd -> distinct banks across rows)

__launch_bounds__(256, 1)
__global__ void ot_prompt_kernel(const float* __restrict__ x,
                                 const float* __restrict__ P,
                                 const int* __restrict__ max_iter_p,
                                 float* __restrict__ out)
{
    __shared__ float xs[NPAD * XS];     // normalized x, padded rows zero
    __shared__ float ps[MPAD * PS];     // normalized P, then raw P (reload)
    __shared__ float Ks[NPAD * KS];     // Gibbs kernel K, then Tn (in place)
    __shared__ float us[NPAD];
    __shared__ float vs[MPAD];
    __shared__ float part[8 * 32];

    const int tid  = threadIdx.x;
    const int lane = tid & 31;
    const int wid  = tid >> 5;
    const int l16  = lane & 15;
    const int half = lane >> 4;         // 0 or 1
    const int koff = half << 1;         // K offset 0 or 2 for WMMA A/B frags

    // ---- Stage 1: load + L2-normalize x and P into LDS (zero padding) ----
    for (int r = wid; r < NPAD; r += 8) {
        if (r < N_FEATS) {
            float v0 = x[r * DIM + lane];
            float v1 = x[r * DIM + lane + 32];
            float v2 = x[r * DIM + lane + 64];
            float v3 = x[r * DIM + lane + 96];
            float s = v0 * v0 + v1 * v1 + v2 * v2 + v3 * v3;
            #pragma unroll
            for (int m = 16; m; m >>= 1) s += __shfl_xor(s, m, 32);
            float rinv = 1.0f / fmaxf(sqrtf(s), 1e-12f);
            xs[r * XS + lane]      = v0 * rinv;
            xs[r * XS + lane + 32] = v1 * rinv;
            xs[r * XS + lane + 64] = v2 * rinv;
            xs[r * XS + lane + 96] = v3 * rinv;
        } else {
            xs[r * XS + lane]      = 0.0f;
            xs[r * XS + lane + 32] = 0.0f;
            xs[r * XS + lane + 64] = 0.0f;
            xs[r * XS + lane + 96] = 0.0f;
        }
    }
    for (int r = wid; r < MPAD; r += 8) {
        if (r < M_PROMPTS) {
            float v0 = P[r * DIM + lane];
            float v1 = P[r * DIM + lane + 32];
            float v2 = P[r * DIM + lane + 64];
            float v3 = P[r * DIM + lane + 96];
            float s = v0 * v0 + v1 * v1 + v2 * v2 + v3 * v3;
            #pragma unroll
            for (int m = 16; m; m >>= 1) s += __shfl_xor(s, m, 32);
            float rinv = 1.0f / fmaxf(sqrtf(s), 1e-12f);
            ps[r * PS + lane]      = v0 * rinv;
            ps[r * PS + lane + 32] = v1 * rinv;
            ps[r * PS + lane + 64] = v2 * rinv;
            ps[r * PS + lane + 96] = v3 * rinv;
        } else {
            ps[r * PS + lane]      = 0.0f;
            ps[r * PS + lane + 32] = 0.0f;
            ps[r * PS + lane + 64] = 0.0f;
            ps[r * PS + lane + 96] = 0.0f;
        }
    }
    __syncthreads();

    // ---- Stage 2: K = exp((x_norm @ p_norm^T - 1)/eps) via f32 WMMA ----
    // 7x2 tiles of 16x16, K-dim 128 in steps of 4. All waves uniform.
    for (int t = wid; t < 14; t += 8) {
        const int tm = t >> 1, tn = t & 1;
        v8f acc = {0.f, 0.f, 0.f, 0.f, 0.f, 0.f, 0.f, 0.f};
        const int arow = tm * 16 + l16;
        const int bcol = tn * 16 + l16;
        #pragma unroll
        for (int kb = 0; kb < DIM; kb += 4) {
            v2f a, b;
            a.x = xs[arow * XS + kb + koff];
            a.y = xs[arow * XS + kb + koff + 1];
            b.x = ps[bcol * PS + kb + koff];      // B[k][n] = p_norm[n][k]
            b.y = ps[bcol * PS + kb + koff + 1];
            acc = __builtin_amdgcn_wmma_f32_16x16x4_f32(
                false, a, false, b, (short)0, acc, false, false);
        }
        #pragma unroll
        for (int g = 0; g < 8; ++g) {
            const int Mr = tm * 16 + g + (half << 3);
            const int Nc = tn * 16 + l16;
            float kv = (Mr < N_FEATS && Nc < M_PROMPTS)
                         ? __expf((acc[g] - 1.0f) * 20.0f)   // -C/eps, eps=0.05
                         : 0.0f;                             // zero padding
            Ks[Mr * KS + Nc] = kv;
        }
    }
    if (tid < MPAD) vs[tid] = (tid < M_PROMPTS) ? (1.0f / M_PROMPTS) : 0.0f;
    __syncthreads();

    // ---- Stage 3: Sinkhorn iterations (deterministic tree reductions) ----
    const int iters = *max_iter_p;
    for (int it = 0; it < iters; ++it) {
        if (tid < NPAD) {
            float s = 0.f;
            #pragma unroll
            for (int j = 0; j < MPAD; ++j) s += Ks[tid * KS + j] * vs[j];
            us[tid] = (tid < N_FEATS) ? (1.0f / N_FEATS) / s : 0.0f;
        }
        __syncthreads();
        {   // partial K^T @ u : wave g sums rows i = g, g+8, ...
            float s = 0.f;
            #pragma unroll
            for (int i = wid; i < NPAD; i += 8) s += Ks[i * KS + lane] * us[i];
            part[wid * 32 + lane] = s;
        }
        __syncthreads();
        if (tid < MPAD) {
            float s = 0.f;
            #pragma unroll
            for (int g = 0; g < 8; ++g) s += part[g * 32 + tid];
            vs[tid] = (tid < M_PROMPTS) ? (1.0f / M_PROMPTS) / s : 0.0f;
        }
        __syncthreads();
    }

    // ---- Stage 4: Tn = N * diag(u) K diag(v), in place; emit Tn output ----
    float* outT = out + N_FEATS * DIM;
    for (int idx = tid; idx < NPAD * MPAD; idx += 256) {   // 14 uniform iters
        const int i = idx >> 5, j = idx & 31;
        float tv = us[i] * Ks[i * KS + j] * vs[j] * (float)N_FEATS;
        Ks[i * KS + j] = tv;
        if (i < N_FEATS && j < M_PROMPTS) outT[i * M_PROMPTS + j] = tv;
    }
    // reload RAW P into ps rows 0..19 (padding rows already zero)
    for (int r = wid; r < M_PROMPTS; r += 8) {
        ps[r * PS + lane]      = P[r * DIM + lane];
        ps[r * PS + lane + 32] = P[r * DIM + lane + 32];
        ps[r * PS + lane + 64] = P[r * DIM + lane + 64];
        ps[r * PS + lane + 96] = P[r * DIM + lane + 96];
    }
    __syncthreads();

    // ---- Stage 5: x_prompted = x + Tn @ P via f32 WMMA (7x8 tiles, K=32) --
    for (int t = wid; t < 56; t += 8) {
        const int tm = t >> 3, tn = t & 7;
        v8f acc = {0.f, 0.f, 0.f, 0.f, 0.f, 0.f, 0.f, 0.f};
        const int arow = tm * 16 + l16;
        const int bcol = tn * 16 + l16;
        #pragma unroll
        for (int kb = 0; kb < MPAD; kb += 4) {
            v2f a, b;
            a.x = Ks[arow * KS + kb + koff];
            a.y = Ks[arow * KS + kb + koff + 1];
            b.x = ps[(kb + koff) * PS + bcol];     // B[k][n] = P[k][n]
            b.y = ps[(kb + koff + 1) * PS + bcol];
            acc = __builtin_amdgcn_wmma_f32_16x16x4_f32(
                false, a, false, b, (short)0, acc, false, false);
        }
        #pragma unroll
        for (int g = 0; g < 8; ++g) {
            const int Mr = tm * 16 + g + (half << 3);
            const int Nc = tn * 16 + l16;
            if (Mr < N_FEATS)
                out[Mr * DIM + Nc] = x[Mr * DIM + Nc] + acc[g];
        }
    }
}

extern "C" void kernel_launch(void* const* d_in, const int* in_sizes, int n_in,
                              void* d_out, int out_size, void* d_ws, size_t ws_size,
                              hipStream_t stream) {
    const float* x        = (const float*)d_in[0];
    const float* P        = (const float*)d_in[1];
    const int*   max_iter = (const int*)d_in[2];
    float*       out      = (float*)d_out;
    (void)in_sizes; (void)n_in; (void)out_size; (void)d_ws; (void)ws_size;
    // Single workgroup: whole problem lives in one WGP's LDS; Sinkhorn's 50
    // dependent iterations sync via workgroup barriers (no global sync).
    ot_prompt_kernel<<<1, 256, 0, stream>>>(x, P, max_iter, out);
}